// ExponentialKernelFiringRateModel_9861244911694
// MI455X (gfx1250) — compile-verified
//
#include <hip/hip_runtime.h>
#include <math.h>
#include <stdint.h>

typedef __attribute__((ext_vector_type(2))) float v2f;
typedef __attribute__((ext_vector_type(4))) float v4f;
typedef __attribute__((ext_vector_type(8))) float v8f;

#define WAVES_PER_BLOCK 8   // 256 threads / wave32
#define CHUNK 32            // timesteps staged per LDS tile
#define HALF_CHUNK 16       // stores re-based every 16 steps (imm offset < 8MB)

#if __has_builtin(__builtin_amdgcn_permlanex16)
#define HAVE_PERMX16 1
#else
#define HAVE_PERMX16 0
#endif

__device__ __forceinline__ float fast_tanh(float x) {
#if __has_builtin(__builtin_amdgcn_tanhf)
    return __builtin_amdgcn_tanhf(x);       // V_TANH_F32 (gfx1250 trans op)
#else
    return tanhf(x);
#endif
}

#if HAVE_PERMX16
// Identity cross-half swap: lane i <- lane (i%16) of the other 16-lane half.
__device__ __forceinline__ float xhalf_swap(float x) {
    unsigned u = __builtin_bit_cast(unsigned, x);
    unsigned r = __builtin_amdgcn_permlanex16(u, u, 0x76543210u, 0xfedcba98u,
                                              false, false);
    return __builtin_bit_cast(float, r);
}
#endif

// One wave32 per batch element. lane owns n = 4*lane .. 4*lane+3 (N=128).
// Scan over T kept entirely in registers; vs stored with NT b128 stores.
__global__ __launch_bounds__(256) void frm_scan_kernel(
    const float* __restrict__ Is,   // [T,B]
    const float* __restrict__ ds,   // [N]
    const float* __restrict__ a,    // [N]
    const float* __restrict__ bvec, // [N]
    const float* __restrict__ w,    // [N]
    const float* __restrict__ pc,   // [4] poly_coeff
    const float* __restrict__ gbp,  // [1]
    float* __restrict__ pred,       // [T,B]
    float* __restrict__ vs,         // [T,B,N]
    int T, int B)
{
    const int N = 128;
    // LDS tile laid out [batch-in-block][time] so each wave's 32 I values
    // are contiguous -> preloaded with 8 uniform ds_load_b128.
    __shared__ float sIs[2][WAVES_PER_BLOCK * CHUNK];

    const int tid  = threadIdx.x;
    const int lane = tid & 31;
    const int wb   = tid >> 5;                      // wave index in block (0..7)
    const int b0   = blockIdx.x * WAVES_PER_BLOCK;  // first batch elem of block
    const int bidx = b0 + wb;                       // this wave's batch element

    // Per-lane constants for its 4 channels.
    v4f ds4 = *(const v4f*)(ds   + 4 * lane);
    v4f a4  = *(const v4f*)(a    + 4 * lane);
    v4f bb4 = *(const v4f*)(bvec + 4 * lane);
    v4f w4  = *(const v4f*)(w    + 4 * lane);
    v4f dm, kb;
    dm.x = 1.0f - ds4.x; dm.y = 1.0f - ds4.y;
    dm.z = 1.0f - ds4.z; dm.w = 1.0f - ds4.w;
    kb.x = 1000.0f * bb4.x; kb.y = 1000.0f * bb4.y;   // fold 1000* into b
    kb.z = 1000.0f * bb4.z; kb.w = 1000.0f * bb4.w;

    // Fold (z - g_b)/1000 into the polynomial: evaluate directly in z.
    const float c0 = pc[0] * pc[0];
    const float c1 = pc[1] * pc[1];
    const float c2 = pc[2] * pc[2];
    const float c3 = pc[3] * pc[3];
    const float s  = 1.0f / 1000.0f;
    const float g  = gbp[0] * s;
    const float d3 = c3 * s * s * s;
    const float d2 = (c2 - 3.0f * c3 * g) * s * s;
    const float d1 = (c1 - 2.0f * c2 * g + 3.0f * c3 * g * g) * s;
    const float d0 = c0 - (c1 - (c2 - c3 * g) * g) * g;

    v4f v = {0.0f, 0.0f, 0.0f, 0.0f};
    float f = 0.0f;

    // All-ones B operand: with B[k][n]=1, WMMA D rows are row-sums of A.
    const v2f Bones = {1.0f, 1.0f};

    float*       vp = vs   + (size_t)bidx * N + 4 * lane;
    float*       pp = pred + bidx;
    const size_t vstep = (size_t)B * N;   // one timestep of vs (floats)

    // One scan step; vs store goes to an explicit address (constant immediate
    // offset from a per-half-chunk base -> no per-step address updates).
    auto step = [&](float I, float* vaddr) {
        // v = (1-ds)*v + (I*a + f*(1000*b)); f-dependent term is depth-1.
        v.x = fmaf(dm.x, v.x, fmaf(f, kb.x, I * a4.x));
        v.y = fmaf(dm.y, v.y, fmaf(f, kb.y, I * a4.y));
        v.z = fmaf(dm.z, v.z, fmaf(f, kb.z, I * a4.z));
        v.w = fmaf(dm.w, v.w, fmaf(f, kb.w, I * a4.w));

        // WMMA A operand = the two pair-dots; row-sums of A still cover all
        // 128 products exactly once (no pre-add, no zero fill needed).
        v2f A1;
        A1[0] = fmaf(v.y, w4.y, v.x * w4.x);
        A1[1] = fmaf(v.w, w4.w, v.z * w4.z);
        v8f acc = {0.0f, 0.0f, 0.0f, 0.0f, 0.0f, 0.0f, 0.0f, 0.0f};
        acc = __builtin_amdgcn_wmma_f32_16x16x4_f32(
            false, A1, false, Bones, (short)0, acc, false, false);
        // lanes 0-15 all hold S_lo = sum(rows 0-7); lanes 16-31 hold S_hi.
        float half = ((acc[0] + acc[1]) + (acc[2] + acc[3])) +
                     ((acc[4] + acc[5]) + (acc[6] + acc[7]));

#if HAVE_PERMX16
        const float z = half + xhalf_swap(half);   // S_lo + S_hi in every lane
#else
        v2f A2; A2[0] = half; A2[1] = 0.0f;
        v8f acc2 = {0.0f, 0.0f, 0.0f, 0.0f, 0.0f, 0.0f, 0.0f, 0.0f};
        acc2 = __builtin_amdgcn_wmma_f32_16x16x4_f32(
            false, A2, false, Bones, (short)0, acc2, false, false);
        const float z = acc2[0];
#endif

        // activation: relu(100 * tanh(poly(z)))  (scale/shift folded into d_i)
        const float poly = fmaf(z, fmaf(z, fmaf(z, d3, d2), d1), d0);
        const float fr   = 100.0f * fast_tanh(poly);
        f = fr > 0.0f ? fr : 0.0f;

        __builtin_nontemporal_store(v, (v4f*)vaddr);
    };

    // Stage one CHUNK x 8-batch tile of Is into LDS with async global->LDS loads.
    auto stage = [&](int c) {
        int tt = tid >> 3;                 // 0..31 timestep within chunk
        int bb = tid & 7;                  // 0..7 batch within block (coalesced)
        int tg = c * CHUNK + tt;
        if (tg > T - 1) tg = T - 1;        // clamp; extra rows never read
        const float* gp = Is + (size_t)tg * B + (b0 + bb);
        unsigned lds_off = (unsigned)(uintptr_t)(&sIs[c & 1][bb * CHUNK + tt]);
        unsigned long long ga = (unsigned long long)(uintptr_t)gp;
        asm volatile("global_load_async_to_lds_b32 %0, %1, off"
                     :: "v"(lds_off), "v"(ga) : "memory");
    };

    stage(0);
    const int nc = (T + CHUNK - 1) / CHUNK;
    for (int c = 0; c < nc; ++c) {
        // Buffer (c+1)&1 was released by the post-preload barrier of chunk c-1,
        // so its refill can start before we even wait for tile c.
        if (c + 1 < nc) {
            stage(c + 1);
            asm volatile("s_wait_asynccnt 0x1" ::: "memory"); // tile c landed
        } else {
            asm volatile("s_wait_asynccnt 0x0" ::: "memory");
        }
        __syncthreads();                    // every wave's tile-c data visible

        const float* sbuf = &sIs[c & 1][wb * CHUNK];
        int tmax = T - c * CHUNK;

        if (tmax >= CHUNK) {
            // Bulk-preload this wave's 32 I values (8 uniform b128 LDS loads),
            // then release the buffer and run 32 register-only steps while the
            // next tile streams in asynchronously.
            v4f Iv[CHUNK / 4];
            #pragma unroll
            for (int k = 0; k < CHUNK / 4; ++k)
                Iv[k] = *(const v4f*)(sbuf + 4 * k);
            __syncthreads();                // buffer free for restaging

            float fv[CHUNK];                // chunk's f outputs (registers)
            #pragma unroll
            for (int h = 0; h < 2; ++h) {   // fixed base per 16 steps:
                #pragma unroll              // imm offsets <= 15*512KB < 8MB
                for (int t2 = 0; t2 < HALF_CHUNK; ++t2) {
                    const int tt = h * HALF_CHUNK + t2;
                    step(Iv[tt >> 2][tt & 3], vp + (size_t)t2 * vstep);
                    fv[tt] = f;
                }
                vp += (size_t)HALF_CHUNK * vstep;
            }

            // One batched pred-store block per chunk (single EXEC toggle,
            // fixed base, small immediate offsets).
            if (lane == 0) {
                #pragma unroll
                for (int tt = 0; tt < CHUNK; ++tt)
                    __builtin_nontemporal_store(fv[tt], pp + (size_t)tt * B);
            }
            pp += (size_t)CHUNK * B;
        } else {
            // Remainder chunk (T % CHUNK steps): simple per-step LDS reads.
            for (int tt = 0; tt < tmax; ++tt) {
                step(sbuf[tt], vp);
                vp += vstep;
                if (lane == 0) __builtin_nontemporal_store(f, pp);
                pp += B;
            }
            __syncthreads();
        }
    }
}

extern "C" void kernel_launch(void* const* d_in, const int* in_sizes, int n_in,
                              void* d_out, int out_size, void* d_ws, size_t ws_size,
                              hipStream_t stream) {
    const float* Is = (const float*)d_in[0];  // [T,B]
    const float* ds = (const float*)d_in[1];  // [N]
    const float* a  = (const float*)d_in[2];  // [N]
    const float* b  = (const float*)d_in[3];  // [N]
    const float* w  = (const float*)d_in[4];  // [N,1]
    const float* pc = (const float*)d_in[5];  // [4]
    const float* gb = (const float*)d_in[6];  // scalar

    const int B = 1024;
    const int T = in_sizes[0] / B;            // in_sizes[0] = T*B

    float* pred = (float*)d_out;              // [T,B] first in return order
    float* vs   = pred + (size_t)T * B;       // then [T,B,N]

    dim3 block(256);                          // 8 waves per block
    dim3 grid(B / WAVES_PER_BLOCK);           // 1024 waves, one per batch elem
    frm_scan_kernel<<<grid, block, 0, stream>>>(Is, ds, a, b, w, pc, gb,
                                                pred, vs, T, B);
}